// MedianFilter_64914135711854
// MI455X (gfx1250) — compile-verified
//
#include <hip/hip_runtime.h>
#include <stdint.h>

// Problem constants (fixed by the reference).
#define KC 3
#define KH 2048
#define KW 2048
// Output tile per block, staged input tile in LDS.
#define TW 64
#define TH 16
#define SW 72   // staged width  = TW + 8 (halo 2 each side, padded to 16B-aligned start)
#define SH 20   // staged height = TH + 4

#if defined(__has_builtin)
#if __has_builtin(__builtin_amdgcn_tensor_load_to_lds) && __has_builtin(__builtin_amdgcn_s_wait_tensorcnt)
#define USE_TDM 1
#endif
#endif

typedef unsigned int u32x4 __attribute__((ext_vector_type(4)));
typedef int          i32x4 __attribute__((ext_vector_type(4)));
typedef int          i32x8 __attribute__((ext_vector_type(8)));

// Compare-exchange: v_min_num_f32 + v_max_num_f32.
__device__ __forceinline__ void ce(float& a, float& b) {
    float lo = __builtin_fminf(a, b);
    float hi = __builtin_fmaxf(a, b);
    a = lo; b = hi;
}

// Exact median-of-25 (13th order statistic), restructured from Devillard's
// 99-CE opt_med25 so that its 21 contiguous sort3 sub-networks lower to
// v_min3_num_f32 / v_med3_num_f32 / v_max3_num_f32 (3 VALU ops instead of 6),
// and the final 3 CEs collapse to a single v_med3_num_f32.
__device__ __forceinline__ float med25(float p[25]) {
#define S2(i, j) ce(p[i], p[j])
#define S3(i, j, k)                                                         \
    {                                                                       \
        float _a = p[i], _b = p[j], _c = p[k];                              \
        p[i] = __builtin_fminf(__builtin_fminf(_a, _b), _c); /* v_min3 */   \
        p[j] = __builtin_amdgcn_fmed3f(_a, _b, _c);          /* v_med3 */   \
        p[k] = __builtin_fmaxf(__builtin_fmaxf(_a, _b), _c); /* v_max3 */   \
    }
    // Phase 1 (independent groups; order within phase is irrelevant).
    S2(0,1);   S3(2,3,4);   S3(5,6,7);   S3(8,9,10);  S3(11,12,13);
    S3(14,15,16); S3(17,18,19); S3(20,21,22); S2(23,24);
    // Phase 2.
    S2(2,5);   S3(0,3,6);   S3(1,4,7);   S3(8,11,14); S3(9,12,15);
    S3(10,13,16); S3(17,20,23); S3(18,21,24); S2(19,22);
    // Phase 3.
    S2(8,17);  S3(0,9,18);  S3(1,10,19); S3(2,11,20); S3(3,12,21);
    S3(4,13,22); S3(5,14,23); S3(6,15,24); S2(7,16);
    // Tail (kept in original CE order; dead halves are DCE'd by the compiler).
    S2(7,19);  S2(13,21); S2(15,23); S2(7,13);  S2(7,15);  S2(1,9);
    S2(3,11);  S2(5,17);  S2(11,17); S2(9,17);  S2(4,10);  S2(6,12);
    S2(7,14);  S2(4,6);   S2(4,7);   S2(12,14); S2(10,14); S2(6,7);
    S2(10,12); S2(6,10);  S2(6,17);  S2(12,17); S2(7,17);  S2(7,10);
    S2(12,18); S2(7,12);  S2(10,18);
    // Original ending s(12,20); s(10,20); s(10,12); return p[12]
    // == median of (p10, p12, p20): one v_med3_num_f32.
    return __builtin_amdgcn_fmed3f(p[10], p[12], p[20]);
#undef S2
#undef S3
}

__global__ __launch_bounds__(256)
void median5x5_kernel(const float* __restrict__ in, float* __restrict__ out) {
    __shared__ __align__(16) float tile[SH * SW];

    const int x0 = blockIdx.x * TW;
    const int y0 = blockIdx.y * TH;
    const int c  = blockIdx.z;
    const float* src = in + (size_t)c * KH * KW;

    const bool interior = (x0 >= 4) && (x0 + TW + 4 <= KW) &&
                          (y0 >= 2) && (y0 + TH + 2 <= KH);

#if defined(USE_TDM)
    if (interior) {
        // Tensor Data Mover: DMA the 72x20 fp32 tile (rows y0-2..y0+17,
        // cols x0-4..x0+67) from global into LDS with one descriptor.
        // Wave 0 issues it (TDM ignores EXEC; other waves skip on EXECZ).
        if (threadIdx.x == 0) {
            const uint64_t gaddr =
                (uint64_t)(uintptr_t)(src + (size_t)(y0 - 2) * KW + (x0 - 4));
            const uint32_t lds = (uint32_t)(uintptr_t)&tile[0];
            // ---- D# group 0 (CDNA5 ISA) ----
            u32x4 g0;
            g0.x = 1u;                                    // count=1, user mode
            g0.y = lds;                                   // lds_addr [63:32]
            g0.z = (uint32_t)gaddr;                       // global_addr [95:64]
            g0.w = (uint32_t)((gaddr >> 32) & 0x1FFFFFFu) // global_addr [120:96]
                 | (2u << 30);                            // type=2 ("image")
            // ---- D# group 1 (§8.4): 2D tensor, 4B elements ----
            i32x8 g1;
            g1.s0 = (int)(2u << 16);                      // wg_mask=0, data_size=2 (4B)
            g1.s1 = (int)(((uint32_t)KW & 0xFFFFu) << 16);          // tensor_dim0 lo16
            g1.s2 = (int)(((uint32_t)KW >> 16) |
                          (((uint32_t)KH & 0xFFFFu) << 16));        // dim0 hi | dim1 lo
            g1.s3 = (int)(((uint32_t)KH >> 16) | ((uint32_t)SW << 16)); // dim1 hi | tile_dim0=72
            g1.s4 = (int)SH;                              // tile_dim1=20, tile_dim2=0
            g1.s5 = (int)KW;                              // tensor_dim0_stride lo32 = 2048
            g1.s6 = 0;                                    // stride0 hi16, stride1 lo16
            g1.s7 = 0;                                    // stride1 hi32
            i32x4 g2 = {0, 0, 0, 0};
            i32x4 g3 = {0, 0, 0, 0};
#if __clang_major__ >= 23
            i32x8 g4 = {0, 0, 0, 0, 0, 0, 0, 0};
            __builtin_amdgcn_tensor_load_to_lds(g0, g1, g2, g3, g4, 0);
#else
            __builtin_amdgcn_tensor_load_to_lds(g0, g1, g2, g3, 0);
#endif
            __builtin_amdgcn_s_wait_tensorcnt(0);         // s_wait_tensorcnt 0
        }
    } else
#endif
    {
        // Border tiles (and non-TDM fallback): clamped scalar staging.
        // TDM returns zeros OOB (wrong for replicate edges), so clamp here.
        for (int idx = threadIdx.x; idx < SH * SW; idx += 256) {
            int r   = idx / SW;
            int col = idx - r * SW;
            int gy = y0 - 2 + r;
            gy = gy < 0 ? 0 : (gy > KH - 1 ? KH - 1 : gy);
            int gx = x0 - 4 + col;
            gx = gx < 0 ? 0 : (gx > KW - 1 ? KW - 1 : gx);
            tile[idx] = src[(size_t)gy * KW + gx];
        }
    }
    __syncthreads();

    // Compute: lanes tx=0..31 all in one output row (conflict-free LDS, fully
    // coalesced stores); wave quad tq=0..7. Each thread: 2 rows x 2 adjacent
    // pixels = 4 independent median networks (ILP for VOPD dual-issue).
    const int tx = threadIdx.x & 31;
    const int tq = threadIdx.x >> 5;
    float* outc = out + (size_t)c * KH * KW;

    for (int rr = 0; rr < 2; ++rr) {
        const int oy   = tq + rr * 8;                 // 0..15 output row in tile
        const int base = oy * SW + (tx * 2 + 2);      // window start (8B aligned)

        // 5x6 register window via float2 LDS loads (ds_load_b64):
        // lane t touches words {2t+2, 2t+3} -> 64 banks covered exactly once.
        float w[5][6];
#pragma unroll
        for (int r = 0; r < 5; ++r) {
#pragma unroll
            for (int cc = 0; cc < 3; ++cc) {
                float2 v = *(const float2*)&tile[base + r * SW + 2 * cc];
                w[r][2 * cc]     = v.x;
                w[r][2 * cc + 1] = v.y;
            }
        }

        // Median is permutation-invariant, so fill p[] column-major with the
        // pixel's UNIQUE column in slots 0..4 and the 4 SHARED columns
        // (identical values for both pixels) in slots 5..24. Phase-1 ops on
        // slots 5..24 are then bit-identical between the two networks and get
        // CSE'd by the compiler.
        float p[25];
#pragma unroll
        for (int r = 0; r < 5; ++r) {
            p[r] = w[r][0];                           // unique col of pixel 0
#pragma unroll
            for (int cc = 1; cc < 5; ++cc)
                p[cc * 5 + r] = w[r][cc];             // shared cols 1..4
        }
        const float m0 = med25(p);

#pragma unroll
        for (int r = 0; r < 5; ++r) {
            p[r] = w[r][5];                           // unique col of pixel 1
#pragma unroll
            for (int cc = 1; cc < 5; ++cc)
                p[cc * 5 + r] = w[r][cc];             // shared cols (same values)
        }
        const float m1 = med25(p);

        const int gy = y0 + oy;
        const int gx = x0 + tx * 2;
        *(float2*)&outc[(size_t)gy * KW + gx] = make_float2(m0, m1);  // global_store_b64
    }
}

extern "C" void kernel_launch(void* const* d_in, const int* in_sizes, int n_in,
                              void* d_out, int out_size, void* d_ws, size_t ws_size,
                              hipStream_t stream) {
    (void)in_sizes; (void)n_in; (void)out_size; (void)d_ws; (void)ws_size;
    const float* image = (const float*)d_in[0];   // [3,2048,2048] fp32
    // d_in[1] is kernel_size == 5 (baked into the kernel).
    float* out = (float*)d_out;                   // [3,2048,2048] fp32
    dim3 grid(KW / TW, KH / TH, KC);              // 32 x 128 x 3 blocks
    median5x5_kernel<<<grid, dim3(256), 0, stream>>>(image, out);
}